// BasicGNN_24086176596429
// MI455X (gfx1250) — compile-verified
//
#include <hip/hip_runtime.h>
#include <cstdint>

#define NN   24
#define HID  64
#define OUTC 65
#define TB   16
#define NPAD 68   // nbuf row stride (floats): 4*nl bank spread -> conflict-free A-frag loads
#define HPAD 20   // hbuf row stride (floats)

typedef __attribute__((ext_vector_type(2))) float v2f;
typedef __attribute__((ext_vector_type(8))) float v8f;

// SMPL tree adjacency (self + parent + children), CSR, compile-time constant.
constexpr int NBR_PTR[NN + 1] = {0,4,7,10,13,16,19,22,25,28,33,35,37,40,43,46,48,51,54,57,60,63,66,68,70};
constexpr int NBR_COL[70] = {
  0,1,2,3,
  0,1,4,   0,2,5,   0,3,6,
  1,4,7,   2,5,8,   3,6,9,
  4,7,10,  5,8,11,
  6,9,12,13,14,
  7,10,    8,11,
  9,12,15, 9,13,16, 9,14,17,
  12,15,
  13,16,18, 14,17,19,
  16,18,20, 17,19,21,
  18,20,22, 19,21,23,
  20,22,   21,23
};

// One 16(batch) x 16(out-ch) tile of the per-node GEMM, K=64, via f32 WMMA.
// All operands are prefetched first (one deep vmem clause + 8 ds_load_2addr_b64),
// then the 16 WMMAs issue back-to-back consuming loads in order.
template<int COUT>
__device__ __forceinline__ void node_gemm_tile(
    const float (&nbuf)[NN][TB][NPAD],
    float (&hbuf)[NN][TB][HPAD],
    const float* __restrict__ W,   // [NN][HID][COUT] row-major
    int node, int jt, int lane)
{
    const int nl = lane & 15;          // A: M (batch row); B/D: N (out channel)
    const int kh = (lane >> 4) << 1;   // K-pair select per half-wave
    const float* wn = W + node * (HID * COUT) + jt * 16 + nl;

    // ---- prefetch all B fragments: 32 dwords per lane, independent loads ----
    v2f b[16];
    #pragma unroll
    for (int ks = 0; ks < 16; ++ks) {
        const int k0 = ks * 4 + kh;
        b[ks].x = wn[k0 * COUT];
        b[ks].y = wn[(k0 + 1) * COUT];
    }
    // ---- prefetch all A fragments: 16 x b64 from LDS (merges to ds_load_2addr) ----
    v2f a[16];
    #pragma unroll
    for (int ks = 0; ks < 16; ++ks) {
        const int k0 = ks * 4 + kh;
        a[ks].x = nbuf[node][nl][k0];
        a[ks].y = nbuf[node][nl][k0 + 1];
    }
    // ---- 16 chained WMMAs ----
    v8f acc = {};
    #pragma unroll
    for (int ks = 0; ks < 16; ++ks)
        acc = __builtin_amdgcn_wmma_f32_16x16x4_f32(false, a[ks], false, b[ks],
                                                    (short)0, acc, false, false);

    const int mh = (lane >> 4) << 3;
    #pragma unroll
    for (int r = 0; r < 8; ++r)
        hbuf[node][mh + r][nl] = acc[r];   // D: vgpr r -> row mh+r, col nl
}

__global__ __launch_bounds__(256)
void gnn4_fused_kernel(const float* __restrict__ x,
                       const float* __restrict__ w0, const float* __restrict__ w1,
                       const float* __restrict__ w2, const float* __restrict__ w3,
                       const float* __restrict__ aw0, const float* __restrict__ aw1,
                       const float* __restrict__ aw2, const float* __restrict__ aw3,
                       const float* __restrict__ b0, const float* __restrict__ b1,
                       const float* __restrict__ b2, const float* __restrict__ b3,
                       float* __restrict__ out)
{
    __shared__ float nbuf[NN][TB][NPAD];   // activations, ~104 KB
    __shared__ float hbuf[NN][TB][HPAD];   // current 16-ch tile of h, ~30 KB
    __shared__ float sAW[NN * NN];
    __shared__ float sBias[OUTC];
    __shared__ float sExtra[NN][TB];       // layer-3 channel 64

    const int tid   = threadIdx.x;
    const int lane  = tid & 31;
    const int wave  = tid >> 5;            // 0..7 (wave32)
    const int bbase = blockIdx.x * TB;

    // ---- load x tile (root joint masked to zero), coalesced float4 ----
    {
        const float4* xb = (const float4*)(x + (size_t)bbase * (NN * HID));
        #pragma unroll
        for (int it = 0; it < 24; ++it) {
            int e4 = tid + it * 256;                 // 6144 float4s total
            float4 v = xb[e4];
            int e = e4 << 2;
            int m = e / (NN * HID);
            int r = e - m * (NN * HID);
            int k = r >> 6;
            int c = r & 63;
            if (k == 0) v = make_float4(0.f, 0.f, 0.f, 0.f);
            *(float4*)&nbuf[k][m][c] = v;
        }
    }

    const int j = tid & 15;   // channel within 16-tile
    const int g = tid >> 4;   // batch row 0..15

    // ---- hidden layers 0..2 ----
    for (int l = 0; l < 3; ++l) {
        const float* W  = (l == 0) ? w0  : (l == 1) ? w1  : w2;
        const float* AW = (l == 0) ? aw0 : (l == 1) ? aw1 : aw2;
        const float* Bi = (l == 0) ? b0  : (l == 1) ? b1  : b2;
        for (int i = tid; i < NN * NN; i += 256) sAW[i] = AW[i];
        if (tid < HID) sBias[tid] = Bi[tid];
        __syncthreads();

        float oreg[96];   // aggregated outputs held in VGPRs until all K-reads done
        #pragma unroll
        for (int jt = 0; jt < 4; ++jt) {
            #pragma unroll
            for (int q = 0; q < 3; ++q)
                node_gemm_tile<HID>(nbuf, hbuf, W, q * 8 + wave, jt, lane);
            __syncthreads();
            #pragma unroll
            for (int n = 0; n < NN; ++n) {          // n compile-time -> scalar sAW reads
                float v = sBias[jt * 16 + j];
                #pragma unroll
                for (int e = NBR_PTR[n]; e < NBR_PTR[n + 1]; ++e)
                    v = fmaf(sAW[n * NN + NBR_COL[e]], hbuf[NBR_COL[e]][g][j], v);
                oreg[jt * 24 + n] = fmaxf(v, 0.0f); // ReLU
            }
            __syncthreads();
        }
        #pragma unroll
        for (int jt = 0; jt < 4; ++jt)
            #pragma unroll
            for (int n = 0; n < NN; ++n)
                nbuf[n][g][jt * 16 + j] = oreg[jt * 24 + n];
        __syncthreads();
    }

    // ---- final layer (65 output channels, no ReLU) ----
    {
        for (int i = tid; i < NN * NN; i += 256) sAW[i] = aw3[i];
        if (tid < OUTC) sBias[tid] = b3[tid];
        // channel 64: per-node length-64 dot in VALU
        for (int p = tid; p < NN * TB; p += 256) {
            int node = p >> 4, m = p & 15;
            const float* wn = w3 + node * (HID * OUTC) + 64;
            float s = 0.f;
            #pragma unroll
            for (int c = 0; c < HID; ++c)
                s = fmaf(nbuf[node][m][c], wn[c * OUTC], s);
            sExtra[node][m] = s;
        }
        __syncthreads();

        float* ob = out + (size_t)bbase * (NN * OUTC);
        #pragma unroll
        for (int jt = 0; jt < 4; ++jt) {
            #pragma unroll
            for (int q = 0; q < 3; ++q)
                node_gemm_tile<OUTC>(nbuf, hbuf, w3, q * 8 + wave, jt, lane);
            __syncthreads();
            #pragma unroll
            for (int n = 0; n < NN; ++n) {
                float v = sBias[jt * 16 + j];
                #pragma unroll
                for (int e = NBR_PTR[n]; e < NBR_PTR[n + 1]; ++e)
                    v = fmaf(sAW[n * NN + NBR_COL[e]], hbuf[NBR_COL[e]][g][j], v);
                ob[(size_t)g * (NN * OUTC) + n * OUTC + jt * 16 + j] = v;
            }
            __syncthreads();
        }
        // channel 64 aggregation
        for (int p = tid; p < NN * TB; p += 256) {
            int n = p >> 4, m = p & 15;
            float v = sBias[64];
            for (int e = NBR_PTR[n]; e < NBR_PTR[n + 1]; ++e)
                v = fmaf(sAW[n * NN + NBR_COL[e]], sExtra[NBR_COL[e]][m], v);
            ob[(size_t)m * (NN * OUTC) + n * OUTC + 64] = v;
        }
    }
}

extern "C" void kernel_launch(void* const* d_in, const int* in_sizes, int n_in,
                              void* d_out, int out_size, void* d_ws, size_t ws_size,
                              hipStream_t stream) {
    const float* x   = (const float*)d_in[0];
    const float* w0  = (const float*)d_in[1];
    const float* w1  = (const float*)d_in[2];
    const float* w2  = (const float*)d_in[3];
    const float* w3  = (const float*)d_in[4];
    const float* aw0 = (const float*)d_in[5];
    const float* aw1 = (const float*)d_in[6];
    const float* aw2 = (const float*)d_in[7];
    const float* aw3 = (const float*)d_in[8];
    const float* b0  = (const float*)d_in[9];
    const float* b1  = (const float*)d_in[10];
    const float* b2  = (const float*)d_in[11];
    const float* b3  = (const float*)d_in[12];
    float* out = (float*)d_out;

    const int blocks = 65536 / TB;   // 4096 workgroups x 256 threads (8 wave32)
    gnn4_fused_kernel<<<blocks, 256, 0, stream>>>(x, w0, w1, w2, w3,
                                                  aw0, aw1, aw2, aw3,
                                                  b0, b1, b2, b3, out);
}